// Attn_32925219291574
// MI455X (gfx1250) — compile-verified
//
#include <hip/hip_runtime.h>

typedef __attribute__((ext_vector_type(2))) float v2f;
typedef __attribute__((ext_vector_type(4))) float v4f;
typedef __attribute__((ext_vector_type(8))) float v8f;

#define HDIM 1024
#define SDIM 4096
#define BDIM 16

// ---------------------------------------------------------------------------
// Kernel 1: uw[j] = sum_k v[k] * W[k, j]   for j in [0, 2*HDIM)
//   uw[0..H)    = w0  (Wh^T v, used for the per-batch constant)
//   uw[H..2H)   = u   (We^T v, used for the streaming dot)
// 8 blocks x 256 threads; consecutive threads read consecutive W columns
// (fully coalesced), v staged in LDS.
// ---------------------------------------------------------------------------
__global__ void __launch_bounds__(256) attn_precompute_uw(
    const float* __restrict__ W, const float* __restrict__ v,
    float* __restrict__ uw) {
  __shared__ float vs[HDIM];
  for (int i = threadIdx.x; i < HDIM; i += 256) vs[i] = v[i];
  __syncthreads();
  const int j = blockIdx.x * 256 + threadIdx.x;   // column of W, 0..2047
  float acc = 0.0f;
#pragma unroll 8
  for (int k = 0; k < HDIM; ++k)
    acc = fmaf(vs[k], W[(size_t)k * (2 * HDIM) + j], acc);
  uw[j] = acc;
}

// ---------------------------------------------------------------------------
// Kernel 2: c[b] = dot(hid[b,:], w0) + dot(v, bias)
// 1 block x 512 threads = 16 waves; wave w handles batch w; shuffle reduce.
// (c[b] is constant over s, so it only shifts the softmax input; kept for
// numerical faithfulness to the reference.)
// ---------------------------------------------------------------------------
__global__ void __launch_bounds__(512) attn_precompute_c(
    const float* __restrict__ hid, const float* __restrict__ bias,
    const float* __restrict__ v, const float* __restrict__ w0,
    float* __restrict__ cbuf) {
  const int wave = threadIdx.x >> 5;   // 0..15 == batch
  const int lane = threadIdx.x & 31;
  float p = 0.0f;
  for (int h = lane; h < HDIM; h += 32) {
    p = fmaf(hid[wave * HDIM + h], w0[h], p);
    p = fmaf(v[h], bias[h], p);
  }
#pragma unroll
  for (int off = 16; off > 0; off >>= 1) p += __shfl_xor(p, off, 32);
  if (lane == 0) cbuf[wave] = p;
}

// ---------------------------------------------------------------------------
// Kernel 3: energies[b,s] = dot(enc[b,s,:], u) + c[b]    (the 256 MB stream)
// One wave per 16-row tile, V_WMMA_F32_16X16X4_F32 accumulating over K=H.
//   A (16x4 f32, from global, NT): lane<16 -> row=lane, K={k0,k0+1};
//                                  lane>=16 -> same rows, K={k0+2,k0+3}
//   B (4x16 f32, from LDS):        u[k] broadcast across all 16 columns;
//                                  same-address reads per lane-half -> LDS
//                                  broadcast, no bank conflicts.
// => every D column holds the 16 dot products; lanes 0/16 extract rows
//    0..7 / 8..15 from their 8 accumulator VGPRs (C/D layout).
// ---------------------------------------------------------------------------
__global__ void __launch_bounds__(256) attn_energies(
    const float* __restrict__ enc, const float* __restrict__ u,
    const float* __restrict__ cbuf, float* __restrict__ out) {
  __shared__ float us[HDIM];
  // Stage u into LDS: 256 threads x 1 b128 load + 1 ds_store_b128.
  {
    const int t4 = threadIdx.x * 4;
    *(v4f*)(us + t4) = *(const v4f*)(u + t4);
  }
  __syncthreads();

  const int wave = threadIdx.x >> 5;
  const int lane = threadIdx.x & 31;
  const int tile = blockIdx.x * 8 + wave;         // 0..4095 (16 rows each)
  const int r0   = tile * 16;                     // global row = b*S + s
  const int m    = lane & 15;                     // row within tile
  const int kb   = (lane >> 4) << 1;              // K sub-offset: 0 or 2

  const float* __restrict__ arow = enc + (size_t)(r0 + m) * HDIM + kb;
  const float* __restrict__ brow = us + kb;       // LDS

  const float cb = cbuf[r0 >> 12];                // S = 4096 rows per batch
  float* __restrict__ outp = out + r0;

#if __has_builtin(__builtin_amdgcn_wmma_f32_16x16x4_f32)
  v8f acc = {};
  for (int kk = 0; kk < HDIM; kk += 64) {
    __builtin_prefetch(arow + kk + 256, 0, 0);    // global_prefetch_b8 -> GL2
#pragma unroll
    for (int k0 = kk; k0 < kk + 64; k0 += 4) {
      // A: single-use stream -> non-temporal global_load_b64
      v2f a  = __builtin_nontemporal_load((const v2f*)(arow + k0));
      // B: LDS broadcast -> ds_load_b64
      v2f bv = *(const v2f*)(brow + k0);
      acc = __builtin_amdgcn_wmma_f32_16x16x4_f32(
          /*neg_a=*/false, a, /*neg_b=*/false, bv,
          /*c_mod=*/(short)0, acc, /*reuse_a=*/false, /*reuse_b=*/false);
    }
  }
  // D: all 16 columns identical; lane 0 holds rows 0..7, lane 16 rows 8..15.
  if (lane == 0) {
#pragma unroll
    for (int j = 0; j < 8; ++j) outp[j] = acc[j] + cb;
  } else if (lane == 16) {
#pragma unroll
    for (int j = 0; j < 8; ++j) outp[8 + j] = acc[j] + cb;
  }
#else
  // Fallback (no f32 WMMA builtin): per-lane partial dot + cross-half shuffle.
  float p = 0.0f;
  for (int kk = 0; kk < HDIM; kk += 64) {
    __builtin_prefetch(arow + kk + 256, 0, 0);
#pragma unroll
    for (int k0 = kk; k0 < kk + 64; k0 += 4) {
      v2f a  = __builtin_nontemporal_load((const v2f*)(arow + k0));
      v2f bv = *(const v2f*)(brow + k0);
      p = fmaf(a.x, bv.x, p);
      p = fmaf(a.y, bv.y, p);
    }
  }
  p += __shfl_xor(p, 16, 32);                     // combine K halves
  if (lane < 16) outp[lane] = p + cb;
#endif
}

// ---------------------------------------------------------------------------
// Kernel 4: in-place softmax over s (4096) per batch row. 16 blocks x 256 thr.
// ---------------------------------------------------------------------------
__global__ void __launch_bounds__(256) attn_softmax(float* __restrict__ out) {
  __shared__ float red[256];
  float* __restrict__ row = out + (size_t)blockIdx.x * SDIM;
  float vals[16];
  float m = -__builtin_inff();
#pragma unroll
  for (int i = 0; i < 16; ++i) {
    vals[i] = row[threadIdx.x + i * 256];
    m = fmaxf(m, vals[i]);
  }
  red[threadIdx.x] = m;
  __syncthreads();
  for (int s = 128; s > 0; s >>= 1) {
    if (threadIdx.x < s)
      red[threadIdx.x] = fmaxf(red[threadIdx.x], red[threadIdx.x + s]);
    __syncthreads();
  }
  m = red[0];
  __syncthreads();
  float sum = 0.0f;
#pragma unroll
  for (int i = 0; i < 16; ++i) {
    vals[i] = expf(vals[i] - m);
    sum += vals[i];
  }
  red[threadIdx.x] = sum;
  __syncthreads();
  for (int s = 128; s > 0; s >>= 1) {
    if (threadIdx.x < s) red[threadIdx.x] += red[threadIdx.x + s];
    __syncthreads();
  }
  const float inv = 1.0f / red[0];
#pragma unroll
  for (int i = 0; i < 16; ++i) row[threadIdx.x + i * 256] = vals[i] * inv;
}

// ---------------------------------------------------------------------------
// Launch: inputs (dict order): hidden, encoder_outputs, W, b, v
// Workspace: [0..1023]=w0, [1024..2047]=u, [2048..2063]=c   (~8.3 KB)
// ---------------------------------------------------------------------------
extern "C" void kernel_launch(void* const* d_in, const int* in_sizes, int n_in,
                              void* d_out, int out_size, void* d_ws, size_t ws_size,
                              hipStream_t stream) {
  (void)in_sizes; (void)n_in; (void)out_size; (void)ws_size;
  const float* hid  = (const float*)d_in[0];   // (16,1,1024)
  const float* enc  = (const float*)d_in[1];   // (16,4096,1024)
  const float* W    = (const float*)d_in[2];   // (1024,2048)
  const float* bias = (const float*)d_in[3];   // (1024,)
  const float* v    = (const float*)d_in[4];   // (1,1024)
  float* out = (float*)d_out;                  // (16,1,4096)
  float* ws  = (float*)d_ws;

  float* w0   = ws;            // [0..1023]
  float* u    = ws + HDIM;     // [1024..2047]
  float* cbuf = ws + 2 * HDIM; // [2048..2063]

  attn_precompute_uw<<<(2 * HDIM) / 256, 256, 0, stream>>>(W, v, ws);
  attn_precompute_c<<<1, 512, 0, stream>>>(hid, bias, v, w0, cbuf);
  attn_energies<<<(BDIM * SDIM / 16) / 8, 256, 0, stream>>>(enc, u, cbuf, out);
  attn_softmax<<<BDIM, 256, 0, stream>>>(out);
}